// GCN_OUTPRODUCT_50165218018132
// MI455X (gfx1250) — compile-verified
//
#include <hip/hip_runtime.h>

typedef __attribute__((ext_vector_type(2))) float v2f;
typedef __attribute__((ext_vector_type(8))) float v8f;

#define Bv 16
#define Cv 32
#define Tv 128
#define Jv 25
#define Ov 64
#define Nv (Tv*Jv)       // 3200 sites per batch
#define C3 (3*Cv)        // 96 concat channels
#define EPSV 1e-5f

#define W3_STRIDE 1026   // padded LDS row stride (floats): mod 64 == 2, 8B-aligned pairs
#define WT_STRIDE 290    // padded LDS row stride (floats) for Wt panel

static __device__ __forceinline__ v8f wmma4(v2f a, v2f b, v8f c) {
  // D(16x16,f32) = A(16x4,f32) x B(4x16,f32) + C  --  V_WMMA_F32_16X16X4_F32
  return __builtin_amdgcn_wmma_f32_16x16x4_f32(false, a, false, b, (short)0, c, false, false);
}

static __device__ __forceinline__ void async_ldsb128(unsigned laddr, unsigned long long gaddr) {
  asm volatile("global_load_async_to_lds_b128 %0, %1, off" :: "v"(laddr), "v"(gaddr) : "memory");
}
static __device__ __forceinline__ void async_ldsb32(unsigned laddr, unsigned long long gaddr) {
  asm volatile("global_load_async_to_lds_b32 %0, %1, off" :: "v"(laddr), "v"(gaddr) : "memory");
}
static __device__ __forceinline__ void wait_async0() {
  asm volatile("s_wait_asynccnt 0x0" ::: "memory");
}

// ---------------- K0: An = D^-1/2 A D^-1/2 ; A2 = diag(dw) - (A - I) ----------------
__global__ void k_prep(const float* __restrict__ A, float* __restrict__ An, float* __restrict__ A2) {
  __shared__ float rsd[Jv];
  __shared__ float dg[Jv];
  int j = threadIdx.x;
  if (j < Jv) {
    float s = 0.f;
    for (int k = 0; k < Jv; ++k) s += A[j*Jv + k];
    rsd[j] = rsqrtf(s);
    dg[j] = s - A[j*Jv + j];   // dw[j] - Aw[j,j] = s - A[j,j]
  }
  __syncthreads();
  if (j < Jv) {
    for (int k = 0; k < Jv; ++k) {
      float a = A[j*Jv + k];
      An[j*Jv + k] = a * rsd[j] * rsd[k];
      A2[j*Jv + k] = (j == k) ? dg[j] : -a;
    }
  }
}

// ---------------- K1: graph convs over J: xg1 = x*An, xg2 = x*A2  ([B,C,T,J]) ----------------
__global__ void k_graph(const float* __restrict__ x, const float* __restrict__ An,
                        const float* __restrict__ A2, float* __restrict__ xg1,
                        float* __restrict__ xg2) {
  __shared__ float sAn[Jv*Jv];
  __shared__ float sA2[Jv*Jv];
  for (int i = threadIdx.x; i < Jv*Jv; i += blockDim.x) { sAn[i] = An[i]; sA2[i] = A2[i]; }
  __syncthreads();
  int idx = blockIdx.x*blockDim.x + threadIdx.x;          // over B*C*T*J
  if (idx >= Bv*Cv*Tv*Jv) return;
  int k   = idx % Jv;
  int row = idx / Jv;                                      // (b*C+c)*T + t
  const float* xr = x + row*Jv;
  float a1 = 0.f, a2 = 0.f;
  #pragma unroll
  for (int jj = 0; jj < Jv; ++jj) {
    float xv = xr[jj];
    a1 += xv * sAn[jj*Jv + k];
    a2 += xv * sA2[jj*Jv + k];
  }
  xg1[idx] = a1;
  xg2[idx] = a2;
}

// ---------------- K2: branches 1&2 pointwise conv via WMMA: relu(W@xg + b) -> xc ----------------
// tiles: branch(2) x b(16) x mt(2) x nt(200) = 12800 waves
__global__ void __launch_bounds__(256) k_pw(
    const float* __restrict__ xg1, const float* __restrict__ xg2,
    const float* __restrict__ W1, const float* __restrict__ b1,
    const float* __restrict__ W2, const float* __restrict__ b2,
    float* __restrict__ xc) {
  int wave = (blockIdx.x*blockDim.x + threadIdx.x) >> 5;
  int lane = threadIdx.x & 31;
  int nt = wave % 200; int r = wave / 200;
  int mt = r & 1; r >>= 1;
  int b  = r & 15; r >>= 4;
  int br = r;                                   // 0: branch1, 1: branch2
  const float* W    = br ? W2 : W1;
  const float* bias = br ? b2 : b1;
  const float* src  = (br ? xg2 : xg1) + b*Cv*Nv;
  int ncol = lane & 15, hi = lane >> 4;
  int n    = nt*16 + ncol;
  int rowA = mt*16 + ncol;
  v8f acc = {0.f,0.f,0.f,0.f,0.f,0.f,0.f,0.f};
  #pragma unroll
  for (int k = 0; k < Cv; k += 4) {
    v2f a, bb;
    a.x  = W[rowA*Cv + k + 2*hi];
    a.y  = W[rowA*Cv + k + 2*hi + 1];
    bb.x = src[(k + 2*hi    )*Nv + n];
    bb.y = src[(k + 2*hi + 1)*Nv + n];
    acc = wmma4(a, bb, acc);
  }
  float* dst = xc + (b*C3 + br*Cv)*Nv;
  #pragma unroll
  for (int rr = 0; rr < 8; ++rr) {
    int m = mt*16 + rr + 8*hi;
    float v = acc[rr] + bias[m];
    dst[m*Nv + n] = v > 0.f ? v : 0.f;
  }
}

// ---------------- K3: branch 3 quadratic form via WMMA -> xc ----------------
// Out[o,n] = sum_{c1,c2} W3[o, c1*32+c2] * v[c1]*v[c2],  v = xg2 column (pre-relu x2r)
// tiles: b(16) x mt(2) x nt(200) = 6400 waves; 256 WMMAs/tile (K=1024)
// All 8 waves of a block share (b,mt) (200 % 8 == 0), so the 16x1024 W3 panel
// is staged once per block into LDS via async global->LDS copies.
__global__ void __launch_bounds__(256) k_outer(
    const float* __restrict__ xg2, const float* __restrict__ W3,
    const float* __restrict__ b3, float* __restrict__ xc) {
  __shared__ float sW3[16 * W3_STRIDE];               // 65,664 B
  int tid  = threadIdx.x;
  int wave = (blockIdx.x*blockDim.x + tid) >> 5;
  int lane = tid & 31;
  int nt = wave % 200; int r = wave / 200;
  int mt = r & 1; int b = r >> 1;

  // ---- stage W3 rows [mt*16, mt*16+16) into padded LDS panel (4096 b128 chunks) ----
  {
    unsigned ldsbase = (unsigned)(uintptr_t)(void*)sW3;
    const char* gbase = (const char*)(W3 + (size_t)mt*16*1024);
    #pragma unroll
    for (int i = 0; i < 16; ++i) {
      int c    = i*256 + tid;                 // chunk id 0..4095
      int rrow = c >> 8;                      // row 0..15
      int col  = (c & 255) * 16;              // byte offset within 4KB row
      unsigned laddr = ldsbase + (unsigned)(rrow*(W3_STRIDE*4) + col);
      unsigned long long gaddr = (unsigned long long)(gbase + (size_t)rrow*4096 + col);
      async_ldsb128(laddr, gaddr);
    }
    wait_async0();
  }
  __syncthreads();

  int ncol = lane & 15, hi = lane >> 4;
  int n    = nt*16 + ncol;
  const float* src = xg2 + b*Cv*Nv;
  float va[Cv];
  #pragma unroll
  for (int c = 0; c < Cv; ++c) va[c] = src[c*Nv + n];
  // per-lane B-operand row values: rows 4*c2b + 2*hi + {0,1}; const indices via select on hi
  float vb[16];
  #pragma unroll
  for (int c2b = 0; c2b < 8; ++c2b) {
    vb[2*c2b]   = hi ? va[4*c2b + 2] : va[4*c2b];
    vb[2*c2b+1] = hi ? va[4*c2b + 3] : va[4*c2b + 1];
  }
  v8f acc = {0.f,0.f,0.f,0.f,0.f,0.f,0.f,0.f};
  int rbase = ncol * W3_STRIDE;               // lane's A row in LDS (panel-local row = ncol)
  #pragma unroll
  for (int c1 = 0; c1 < Cv; ++c1) {
    float vc1 = va[c1];
    #pragma unroll
    for (int c2b = 0; c2b < 8; ++c2b) {
      int kk = c1*Cv + 4*c2b;
      v2f a, bb;
      a.x  = sW3[rbase + kk + 2*hi];
      a.y  = sW3[rbase + kk + 2*hi + 1];
      bb.x = vc1 * vb[2*c2b];
      bb.y = vc1 * vb[2*c2b + 1];
      acc = wmma4(a, bb, acc);
    }
  }
  int rowA = mt*16 + ncol; (void)rowA;
  float* dst = xc + (b*C3 + 2*Cv)*Nv;
  #pragma unroll
  for (int rr = 0; rr < 8; ++rr) {
    int m = mt*16 + rr + 8*hi;
    float v = acc[rr] + b3[m];
    dst[m*Nv + n] = v > 0.f ? v : 0.f;
  }
}

// ---------------- K4: temporal conv (3,1) pad (1,0) via WMMA + bias + relu -> y [B,O,T,J] ----------------
// y[o,n] = sum_dt sum_i Wt[o,i,dt+1] * xc[i, n + 25*dt] (masked at t edges)
// tiles: b(16) x mt(4) x nt(200) = 12800 waves; K = 96 per dt.
// Block-uniform (b,mt): stage the 16x288 Wt panel once per block into LDS (async).
__global__ void __launch_bounds__(256) k_tconv(
    const float* __restrict__ xc, const float* __restrict__ Wt,
    const float* __restrict__ bt, float* __restrict__ y) {
  __shared__ float sWt[16 * WT_STRIDE];               // 18,560 B
  int tid  = threadIdx.x;
  int wave = (blockIdx.x*blockDim.x + tid) >> 5;
  int lane = tid & 31;
  int nt = wave % 200; int r = wave / 200;
  int mt = r & 3; int b = r >> 2;

  // ---- stage Wt rows [mt*16, mt*16+16) (each 288 floats, 4608 total) ----
  {
    unsigned ldsbase = (unsigned)(uintptr_t)(void*)sWt;
    const char* gbase = (const char*)(Wt + (size_t)mt*16*288);
    #pragma unroll
    for (int i = 0; i < 18; ++i) {
      int c    = i*256 + tid;                 // float id 0..4607
      int rrow = c / 288;
      int col  = c % 288;
      unsigned laddr = ldsbase + (unsigned)((rrow*WT_STRIDE + col)*4);
      unsigned long long gaddr = (unsigned long long)(gbase + (size_t)(rrow*288 + col)*4);
      async_ldsb32(laddr, gaddr);
    }
    wait_async0();
  }
  __syncthreads();

  int ncol = lane & 15, hi = lane >> 4;
  int n    = nt*16 + ncol;
  const float* xcb = xc + b*C3*Nv;
  int arow = ncol * WT_STRIDE;                // lane's A row in LDS (panel-local row = ncol)
  v8f acc = {0.f,0.f,0.f,0.f,0.f,0.f,0.f,0.f};
  #pragma unroll
  for (int dt = -1; dt <= 1; ++dt) {
    int np = n + Jv*dt;
    bool valid = (np >= 0) && (np < Nv);
    int rr = dt + 1;
    #pragma unroll
    for (int k = 0; k < C3; k += 4) {
      v2f a, bb;
      a.x  = sWt[arow + (k + 2*hi    )*3 + rr];
      a.y  = sWt[arow + (k + 2*hi + 1)*3 + rr];
      bb.x = valid ? xcb[(k + 2*hi    )*Nv + np] : 0.f;
      bb.y = valid ? xcb[(k + 2*hi + 1)*Nv + np] : 0.f;
      acc = wmma4(a, bb, acc);
    }
  }
  #pragma unroll
  for (int rr = 0; rr < 8; ++rr) {
    int o = mt*16 + rr + 8*hi;
    float v = acc[rr] + bt[o];
    y[(b*Ov + o)*Nv + n] = v > 0.f ? v : 0.f;
  }
}

// ---------------- K5: BN stats per channel -> scale/shift ----------------
__global__ void k_bnstats(const float* __restrict__ y, const float* __restrict__ gamma,
                          const float* __restrict__ beta, float* __restrict__ scale,
                          float* __restrict__ shift) {
  int o = blockIdx.x;                       // 64 channels
  float s = 0.f, s2 = 0.f;
  for (int i = threadIdx.x; i < Bv*Nv; i += blockDim.x) {
    int bb = i / Nv, n = i % Nv;
    float v = y[(bb*Ov + o)*Nv + n];
    s += v; s2 += v*v;
  }
  __shared__ float sh[256];
  __shared__ float sh2[256];
  sh[threadIdx.x] = s; sh2[threadIdx.x] = s2;
  __syncthreads();
  for (int st = 128; st > 0; st >>= 1) {
    if (threadIdx.x < st) { sh[threadIdx.x] += sh[threadIdx.x+st]; sh2[threadIdx.x] += sh2[threadIdx.x+st]; }
    __syncthreads();
  }
  if (threadIdx.x == 0) {
    const float inv = 1.0f / (float)(Bv*Nv);
    float mean = sh[0]*inv;
    float var  = sh2[0]*inv - mean*mean;
    float sc   = gamma[o]*rsqrtf(var + EPSV);
    scale[o] = sc;
    shift[o] = beta[o] - mean*sc;
  }
}

// ---------------- K6: apply BN -> d_out ----------------
__global__ void k_bnapply(const float* __restrict__ y, const float* __restrict__ scale,
                          const float* __restrict__ shift, float* __restrict__ out) {
  int idx = blockIdx.x*blockDim.x + threadIdx.x;
  if (idx >= Bv*Ov*Nv) return;
  int o = (idx / Nv) % Ov;
  out[idx] = y[idx]*scale[o] + shift[o];
}

extern "C" void kernel_launch(void* const* d_in, const int* in_sizes, int n_in,
                              void* d_out, int out_size, void* d_ws, size_t ws_size,
                              hipStream_t stream) {
  const float* x     = (const float*)d_in[0];
  const float* A     = (const float*)d_in[1];
  const float* W1    = (const float*)d_in[2];
  const float* b1    = (const float*)d_in[3];
  const float* W2    = (const float*)d_in[4];
  const float* b2    = (const float*)d_in[5];
  const float* W3    = (const float*)d_in[6];
  const float* b3    = (const float*)d_in[7];
  const float* Wt    = (const float*)d_in[8];
  const float* bt    = (const float*)d_in[9];
  const float* gamma = (const float*)d_in[10];
  const float* beta  = (const float*)d_in[11];

  float* ws    = (float*)d_ws;
  float* An    = ws + 0;                    // 625
  float* A2m   = ws + 640;                  // 625
  float* scale = ws + 1280;                 // 64
  float* shift = ws + 1344;                 // 64
  float* xcb   = ws + 4096;                 // B*96*N  = 4,915,200
  float* xg1   = xcb + (size_t)Bv*C3*Nv;    // B*32*N  = 1,638,400
  float* xg2   = xg1 + (size_t)Bv*Cv*Nv;    // B*32*N  = 1,638,400
  float* yb    = xg2 + (size_t)Bv*Cv*Nv;    // B*64*N  = 3,276,800

  k_prep   <<<1,     32, 0, stream>>>(A, An, A2m);
  k_graph  <<<6400, 256, 0, stream>>>(x, An, A2m, xg1, xg2);
  k_pw     <<<1600, 256, 0, stream>>>(xg1, xg2, W1, b1, W2, b2, xcb);   // 12800 waves
  k_outer  <<<800,  256, 0, stream>>>(xg2, W3, b3, xcb);                // 6400 waves
  k_tconv  <<<1600, 256, 0, stream>>>(xcb, Wt, bt, yb);                 // 12800 waves
  k_bnstats<<<64,   256, 0, stream>>>(yb, gamma, beta, scale, shift);
  k_bnapply<<<12800,256, 0, stream>>>(yb, scale, shift, (float*)d_out);
}